// HierarchicalGraphPooling_78142634983585
// MI455X (gfx1250) — compile-verified
//
#include <hip/hip_runtime.h>

// Problem constants (match reference setup_inputs()).
#define N_NODES   100000
#define N_EDGES   3200000
#define DIM       256
#define K_POOL    50000                 // int(100000 * 0.5)
#define SORT_TILE 1024                  // elements per sort block
#define NUM_SORT_BLOCKS ((N_NODES + SORT_TILE - 1) / SORT_TILE)   // 98
#define HIST_SIZE (256 * NUM_SORT_BLOCKS)

typedef __attribute__((ext_vector_type(2))) float v2f;
typedef __attribute__((ext_vector_type(8))) float v8f;

// Monotone map f32 -> u32 (ascending), then invert so that an ASCENDING
// stable radix sort on the key yields DESCENDING scores with lowest-index
// tie-break — exactly jax.lax.top_k ordering over the full array.
__device__ __forceinline__ unsigned int score_key(float s) {
  unsigned int u = __float_as_uint(s);
  u = (u & 0x80000000u) ? ~u : (u | 0x80000000u);
  return ~u;
}

// -------- Stage 1: scores = x @ W + b via V_WMMA_F32_16X16X4_F32 --------
// One wave32 computes 16 rows. A (16x4 f32): lane layout per ISA — lanes
// 0-15 hold K=0,1 in VGPRs 0,1; lanes 16-31 hold K=2,3. B (4x16) carries
// W[k..k+3] replicated across all 16 columns (same lane striping), so every
// column of the 16x16 f32 accumulator converges to the exact row dot-product.
__global__ void __launch_bounds__(256)
scores_wmma_kernel(const float* __restrict__ x,
                   const float* __restrict__ W,
                   const float* __restrict__ b,
                   unsigned int* __restrict__ keys,
                   unsigned int* __restrict__ vals) {
  const int lane    = threadIdx.x & 31;
  const int wave    = threadIdx.x >> 5;
  const int rowBase = (blockIdx.x * 8 + wave) * 16;
  if (rowBase >= N_NODES) return;          // uniform per-wave guard (EXEC stays all-1)

  const int grp = lane >> 4;               // 0: K 0/1 ; 1: K 2/3
  const int l   = lane & 15;
  const float* xrow = x + (size_t)(rowBase + l) * DIM;

  v8f acc = {};
  #pragma unroll 4
  for (int k = 0; k < DIM; k += 4) {
    v2f a = *reinterpret_cast<const v2f*>(xrow + k + 2 * grp);   // 8B aligned
    v2f w = *reinterpret_cast<const v2f*>(W    + k + 2 * grp);
    acc = __builtin_amdgcn_wmma_f32_16x16x4_f32(
        /*neg_a=*/false, a, /*neg_b=*/false, w,
        /*c_mod=*/(short)0, acc, /*reuse_a=*/false, /*reuse_b=*/false);
  }

  // D layout: lanes 0-15 hold M=v (VGPR v), lanes 16-31 hold M=v+8.
  if (l == 0) {
    const float bb = b[0];
    const int r0 = rowBase + grp * 8;
    #pragma unroll
    for (int v = 0; v < 8; ++v) {
      keys[r0 + v] = score_key(acc[v] + bb);
      vals[r0 + v] = (unsigned int)(r0 + v);
    }
  }
}

// -------- Stage 2: stable LSD radix sort (4 x 8-bit passes) --------
__global__ void __launch_bounds__(256)
radix_hist_kernel(const unsigned int* __restrict__ keys, int n, int shift,
                  unsigned int* __restrict__ hist) {
  __shared__ unsigned int h[256];
  h[threadIdx.x] = 0;
  __syncthreads();
  const int base = blockIdx.x * SORT_TILE;
  for (int i = threadIdx.x; i < SORT_TILE; i += 256) {
    const int idx = base + i;
    if (idx < n) atomicAdd(&h[(keys[idx] >> shift) & 255u], 1u);
  }
  __syncthreads();
  // [digit][block] layout so a flat exclusive scan gives stable scatter bases.
  hist[threadIdx.x * NUM_SORT_BLOCKS + blockIdx.x] = h[threadIdx.x];
}

__global__ void __launch_bounds__(256)
radix_scan_kernel(unsigned int* __restrict__ hist) {
  __shared__ unsigned int sums[256];
  const int chunk = (HIST_SIZE + 255) / 256;      // 98
  const int beg = threadIdx.x * chunk;
  const int end = min(beg + chunk, HIST_SIZE);
  unsigned int s = 0;
  for (int i = beg; i < end; ++i) s += hist[i];
  sums[threadIdx.x] = s;
  __syncthreads();
  if (threadIdx.x == 0) {
    unsigned int run = 0;
    for (int i = 0; i < 256; ++i) { unsigned int t = sums[i]; sums[i] = run; run += t; }
  }
  __syncthreads();
  unsigned int run = sums[threadIdx.x];
  for (int i = beg; i < end; ++i) { unsigned int t = hist[i]; hist[i] = run; run += t; }
}

__global__ void __launch_bounds__(256)
radix_scatter_kernel(const unsigned int* __restrict__ keysIn,
                     const unsigned int* __restrict__ valsIn,
                     unsigned int* __restrict__ keysOut,
                     unsigned int* __restrict__ valsOut,
                     int n, int shift, const unsigned int* __restrict__ hist) {
  __shared__ unsigned int sKey[SORT_TILE];
  __shared__ unsigned int sVal[SORT_TILE];
  __shared__ unsigned int sRank[SORT_TILE];
  __shared__ unsigned int cnt[256];
  __shared__ unsigned int base[256];

  const int blockStart = blockIdx.x * SORT_TILE;
  const int m = min(SORT_TILE, n - blockStart);
  for (int i = threadIdx.x; i < m; i += 256) {
    sKey[i] = keysIn[blockStart + i];
    sVal[i] = valsIn[blockStart + i];
  }
  cnt[threadIdx.x]  = 0;
  base[threadIdx.x] = hist[threadIdx.x * NUM_SORT_BLOCKS + blockIdx.x];
  __syncthreads();
  if (threadIdx.x == 0) {                 // serial, but guarantees stability
    for (int i = 0; i < m; ++i) {
      const unsigned int d = (sKey[i] >> shift) & 255u;
      sRank[i] = cnt[d]++;
    }
  }
  __syncthreads();
  for (int i = threadIdx.x; i < m; i += 256) {
    const unsigned int d = (sKey[i] >> shift) & 255u;
    const unsigned int pos = base[d] + sRank[i];
    keysOut[pos] = sKey[i];
    valsOut[pos] = sVal[i];
  }
}

// -------- Stage 3: node remap table --------
__global__ void __launch_bounds__(256)
init_map_kernel(int* __restrict__ node_map) {
  const int i = blockIdx.x * blockDim.x + threadIdx.x;
  if (i < N_NODES) node_map[i] = -1;
}

__global__ void __launch_bounds__(256)
set_map_kernel(const unsigned int* __restrict__ top_idx, int* __restrict__ node_map) {
  const int r = blockIdx.x * blockDim.x + threadIdx.x;
  if (r < K_POOL) node_map[top_idx[r]] = r;
}

// -------- Stage 4: gather pooled_x (float4 streams) + pooled_batch --------
__global__ void __launch_bounds__(64)
gather_kernel(const float* __restrict__ x,
              const long long* __restrict__ batch,
              const unsigned int* __restrict__ top_idx,
              float* __restrict__ outX, float* __restrict__ outBatch) {
  const int r = blockIdx.x;
  const unsigned int src = top_idx[r];
  const float4* xs = reinterpret_cast<const float4*>(x + (size_t)src * DIM);
  float4* xd       = reinterpret_cast<float4*>(outX + (size_t)r * DIM);
  xd[threadIdx.x] = xs[threadIdx.x];        // 64 lanes x float4 = 256 floats
  if (threadIdx.x == 0) outBatch[r] = (float)batch[src];
}

// -------- Stage 5: edge remap + validity --------
__global__ void __launch_bounds__(256)
edges_kernel(const long long* __restrict__ edge_index,
             const int* __restrict__ node_map,
             float* __restrict__ outEdges, float* __restrict__ outValid) {
  const int e = blockIdx.x * blockDim.x + threadIdx.x;
  if (e >= N_EDGES) return;
  const long long s = edge_index[e];
  const long long t = edge_index[(size_t)N_EDGES + e];
  const int ms = node_map[s];
  const int mt = node_map[t];
  const bool valid = (ms >= 0) && (mt >= 0);
  outEdges[e]                      = valid ? (float)ms : -1.0f;
  outEdges[(size_t)N_EDGES + e]    = valid ? (float)mt : -1.0f;
  outValid[e] = valid ? 1.0f : 0.0f;
}

extern "C" void kernel_launch(void* const* d_in, const int* in_sizes, int n_in,
                              void* d_out, int out_size, void* d_ws, size_t ws_size,
                              hipStream_t stream) {
  const float*     x          = (const float*)d_in[0];
  const long long* edge_index = (const long long*)d_in[1];   // int64 per reference
  const long long* batch      = (const long long*)d_in[2];   // int64 per reference
  const float*     W          = (const float*)d_in[3];
  const float*     b          = (const float*)d_in[4];

  // Workspace carve-up (~2.1 MB).
  unsigned int* keys0 = (unsigned int*)d_ws;
  unsigned int* vals0 = keys0 + N_NODES;
  unsigned int* keys1 = vals0 + N_NODES;
  unsigned int* vals1 = keys1 + N_NODES;
  unsigned int* hist  = vals1 + N_NODES;               // HIST_SIZE u32
  int*          node_map = (int*)(hist + HIST_SIZE);   // N_NODES i32

  // Output carve-up (flat f32, tuple order).
  float* outX     = (float*)d_out;                       // K_POOL * DIM
  float* outEdges = outX + (size_t)K_POOL * DIM;         // 2 * N_EDGES
  float* outBatch = outEdges + 2 * (size_t)N_EDGES;      // K_POOL
  float* outValid = outBatch + K_POOL;                   // N_EDGES

  // 1) scores -> sortable (key, node) pairs. 6250 waves of 16 rows.
  const int scoreBlocks = (N_NODES / 16 + 7) / 8;        // 782 blocks x 8 waves
  scores_wmma_kernel<<<scoreBlocks, 256, 0, stream>>>(x, W, b, keys0, vals0);

  // 2) 4-pass stable LSD radix sort (ascending key == descending score).
  unsigned int* kin = keys0; unsigned int* vin = vals0;
  unsigned int* kout = keys1; unsigned int* vout = vals1;
  for (int pass = 0; pass < 4; ++pass) {
    const int shift = pass * 8;
    radix_hist_kernel<<<NUM_SORT_BLOCKS, 256, 0, stream>>>(kin, N_NODES, shift, hist);
    radix_scan_kernel<<<1, 256, 0, stream>>>(hist);
    radix_scatter_kernel<<<NUM_SORT_BLOCKS, 256, 0, stream>>>(kin, vin, kout, vout,
                                                              N_NODES, shift, hist);
    unsigned int* tk = kin; kin = kout; kout = tk;
    unsigned int* tv = vin; vin = vout; vout = tv;
  }
  const unsigned int* top_idx = vin;   // after 4 passes: sorted order lives in keys0/vals0

  // 3) node_map: -1 everywhere, rank r at top_idx[r].
  init_map_kernel<<<(N_NODES + 255) / 256, 256, 0, stream>>>(node_map);
  set_map_kernel<<<(K_POOL + 255) / 256, 256, 0, stream>>>(top_idx, node_map);

  // 4) pooled_x / pooled_batch gather.
  gather_kernel<<<K_POOL, 64, 0, stream>>>(x, batch, top_idx, outX, outBatch);

  // 5) edge remap + validity.
  edges_kernel<<<(N_EDGES + 255) / 256, 256, 0, stream>>>(edge_index, node_map,
                                                          outEdges, outValid);
}